// PredictionLayer_23785528885486
// MI455X (gfx1250) — compile-verified
//
#include <hip/hip_runtime.h>

// Problem constants (match reference)
#define DIM      128
#define HID      64
#define NB       16
#define KN       64
#define KH       128
#define NFIELDS  60082
#define MROWS    (NB * KH)   // 2048 gathered his rows

// Workspace layout (float offsets)
#define WS_V      0          // 128: v = Wp @ w_field
#define WS_WH     128        // 64:  wh = W2 @ w_field
#define WS_CC     192        // 16:  cc[b] = company_out[b] + b_field
#define WS_BPW    208        // 1:   bp . w_field
#define WS_C2     209        // 1:   b2 . w_field
#define WS_DBASE  256        // 60082: field_table[n] . v
#define WS_EHIS   60352      // 2048: per-(b,his-slot) mlp-dot-w_field
// total ~62400 floats (~250 KB)

typedef __attribute__((ext_vector_type(2))) float v2f;
typedef __attribute__((ext_vector_type(8))) float v8f;

// ---------------------------------------------------------------------------
// Kernel 1: tiny setup — fold everything batch-independent into small vectors
// ---------------------------------------------------------------------------
__global__ void __launch_bounds__(128) setup_kernel(
    const float* __restrict__ company_emb, const int* __restrict__ com_id,
    const float* __restrict__ company_table, const float* __restrict__ Wp,
    const float* __restrict__ bp, const float* __restrict__ theta,
    const float* __restrict__ W2, const float* __restrict__ b2,
    const float* __restrict__ w_field, const float* __restrict__ b_field,
    const float* __restrict__ w_company, const float* __restrict__ b_company,
    float* __restrict__ ws) {
  const int t = threadIdx.x;  // 0..127
  // v[k] = sum_d Wp[k][d] * w_field[d]
  {
    float acc = 0.f;
    for (int d = 0; d < DIM; ++d) acc += Wp[t * DIM + d] * w_field[d];
    ws[WS_V + t] = acc;
  }
  if (t < HID) {  // wh[j] = sum_d W2[j][d] * w_field[d]
    float acc = 0.f;
    for (int d = 0; d < DIM; ++d) acc += W2[t * DIM + d] * w_field[d];
    ws[WS_WH + t] = acc;
  }
  if (t == 0) {
    float acc = 0.f;
    for (int d = 0; d < DIM; ++d) acc += bp[d] * w_field[d];
    ws[WS_BPW] = acc;
  }
  if (t == 1) {
    float acc = 0.f;
    for (int d = 0; d < DIM; ++d) acc += b2[d] * w_field[d];
    ws[WS_C2] = acc;
  }
  if (t < NB) {  // cc[b] = company_mem[b] . w_company + b_company + b_field
    const int cid = com_id[t];
    const float th = theta[cid];
    float acc = 0.f;
    for (int d = 0; d < DIM; ++d) {
      const float cm = (1.f - th) * company_emb[t * DIM + d] +
                       th * company_table[(size_t)cid * DIM + d];
      acc += cm * w_company[d];
    }
    ws[WS_CC + t] = acc + b_company[0] + b_field[0];
  }
}

// ---------------------------------------------------------------------------
// Kernel 2: his-MLP hidden layer via V_WMMA_F32_16X16X4_F32.
//   H = raw[his] @ W1   (2048 x 128) @ (128 x 64), fp32 WMMA, K=128 in 32 steps.
//   Epilogue fuses +b1, leaky_relu(0.01), *wh[j], and the per-row column sum
//   (16-lane butterfly) -> e_his[m] = lrelu(raw[m]@W1+b1) . wh + c2.
// One block = 16 rows; 4 waves cover the 4 column tiles (HID=64).
// ---------------------------------------------------------------------------
__global__ void __launch_bounds__(128) mlp_wmma_kernel(
    const float* __restrict__ raw_field_embed, const int* __restrict__ his_nodes,
    const float* __restrict__ W1, const float* __restrict__ b1,
    float* __restrict__ ws) {
  __shared__ float w1s[DIM * HID];  // 32 KB (of 320 KB/WGP)
  __shared__ float b1s[HID];
  __shared__ float whs[HID];
  __shared__ float e_s[16];

  const int tid = threadIdx.x;  // 0..127
  for (int i = tid; i < DIM * HID; i += 128) w1s[i] = W1[i];
  if (tid < HID) { b1s[tid] = b1[tid]; whs[tid] = ws[WS_WH + tid]; }
  if (tid < 16) e_s[tid] = 0.f;
  __syncthreads();

  const int wave = tid >> 5;        // 0..3  -> column tile
  const int lane = tid & 31;
  const int mt = blockIdx.x;        // 0..127 -> 16-row tile
  const int half = lane >> 4;       // 0/1: K-subgroup per ISA f32 A/B layout
  const int nn = lane & 15;         // row-within-tile (A) / col-within-tile (B)
  const int jcol = wave * 16 + nn;  // hidden unit index 0..63

  // gather: row m = mt*16 + nn maps directly into flattened his_nodes[B*KH]
  const int m = (mt << 4) + nn;
  const int node = his_nodes[m];
  const float* __restrict__ arow = raw_field_embed + (size_t)node * DIM;

  v8f c = {};
  for (int kk = 0; kk < DIM; kk += 4) {
    const int k = kk + 2 * half;  // lanes 0-15: K={kk,kk+1}; lanes 16-31: K={kk+2,kk+3}
    v2f a;  a.x = arow[k];                 a.y = arow[k + 1];
    v2f bb; bb.x = w1s[k * HID + jcol];    bb.y = w1s[(k + 1) * HID + jcol];
    c = __builtin_amdgcn_wmma_f32_16x16x4_f32(
        /*neg_a=*/false, a, /*neg_b=*/false, bb,
        /*c_mod=*/(short)0, c, /*reuse_a=*/false, /*reuse_b=*/false);
  }

  // C layout: lane holds H[M = v + 8*half][jcol] in c[v], v = 0..7
  const float wj = whs[jcol];
  const float bj = b1s[jcol];
  float sums[8];
#pragma unroll
  for (int v = 0; v < 8; ++v) {
    float x = c[v] + bj;
    x = (x > 0.f) ? x : 0.01f * x;  // jax.nn.leaky_relu default slope
    x *= wj;
    // butterfly sum over the 16 lanes sharing this M (bits 0..3 stay in-half)
    x += __shfl_xor(x, 1, 32);
    x += __shfl_xor(x, 2, 32);
    x += __shfl_xor(x, 4, 32);
    x += __shfl_xor(x, 8, 32);
    sums[v] = x;
  }
  if ((lane & 15) == 0) {
#pragma unroll
    for (int v = 0; v < 8; ++v) atomicAdd(&e_s[v + 8 * half], sums[v]);
  }
  __syncthreads();
  if (tid < 16) ws[WS_EHIS + (mt << 4) + tid] = e_s[tid] + ws[WS_C2];
}

// ---------------------------------------------------------------------------
// Kernel 3: the bandwidth carrier — d_base[n] = field_table[n] . v
// One wave per row: 32 lanes x float4 = 512B fully-coalesced row read.
// ---------------------------------------------------------------------------
__global__ void __launch_bounds__(256) basedot_kernel(
    const float* __restrict__ field_table, float* __restrict__ ws) {
  const int wave = blockIdx.x * (blockDim.x >> 5) + (threadIdx.x >> 5);
  const int lane = threadIdx.x & 31;
  if (wave >= NFIELDS) return;  // uniform per wave
  const float4* __restrict__ row =
      (const float4*)(field_table + (size_t)wave * DIM);
  const float4* __restrict__ v4 = (const float4*)(ws + WS_V);
  const float4 a = row[lane];
  const float4 vv = v4[lane];
  float x = a.x * vv.x + a.y * vv.y + a.z * vv.z + a.w * vv.w;
  x += __shfl_xor(x, 1, 32);
  x += __shfl_xor(x, 2, 32);
  x += __shfl_xor(x, 4, 32);
  x += __shfl_xor(x, 8, 32);
  x += __shfl_xor(x, 16, 32);
  if (lane == 0) ws[WS_DBASE + wave] = x;
}

// ---------------------------------------------------------------------------
// Kernel 4: write the full output: out[b,n] = d_base[n] + bpw + cc[b]
// Thread per n; 16 coalesced stores per thread. d_base (240 KB) stays in L2.
// ---------------------------------------------------------------------------
__global__ void __launch_bounds__(256) broadcast_kernel(
    const float* __restrict__ ws, float* __restrict__ out) {
  __shared__ float ccs[NB];
  __shared__ float bpws;
  if (threadIdx.x < NB) ccs[threadIdx.x] = ws[WS_CC + threadIdx.x];
  if (threadIdx.x == NB) bpws = ws[WS_BPW];
  __syncthreads();
  const int n = blockIdx.x * blockDim.x + threadIdx.x;
  if (n >= NFIELDS) return;
  const float db = ws[WS_DBASE + n] + bpws;
#pragma unroll
  for (int b = 0; b < NB; ++b) out[(size_t)b * NFIELDS + n] = db + ccs[b];
}

// ---------------------------------------------------------------------------
// Kernel 5: correction pass, one block per batch row. Each slot computes the
// FINAL value for its node (checking membership in both lists), so duplicate
// writers store bitwise-identical values (matches jax .set semantics).
// ---------------------------------------------------------------------------
__global__ void __launch_bounds__(192) correction_kernel(
    const float* __restrict__ field_emb, const int* __restrict__ now_nodes,
    const int* __restrict__ his_nodes, const float* __restrict__ alpha_fields,
    const float* __restrict__ w_field, const float* __restrict__ ws,
    float* __restrict__ out) {
  __shared__ int nowL[KN];
  __shared__ int hisL[KH];
  __shared__ float wfs[DIM];
  const int b = blockIdx.x;
  const int t = threadIdx.x;  // 0..191
  if (t < KN) nowL[t] = now_nodes[b * KN + t];
  if (t < KH) hisL[t] = his_nodes[b * KH + t];
  if (t < DIM) wfs[t] = w_field[t];
  __syncthreads();

  const int n = (t < KN) ? nowL[t] : hisL[t - KN];
  bool isNow = false;
  for (int i = 0; i < KN; ++i) isNow |= (nowL[i] == n);
  int hisIdx = -1;
  for (int i = 0; i < KH; ++i)
    if (hisL[i] == n && hisIdx < 0) hisIdx = i;

  const float alpha = alpha_fields[n];
  const float db = ws[WS_DBASE + n] + ws[WS_BPW];

  float nowE = 0.f;
  if (isNow) {
    const float* __restrict__ fe = field_emb + (size_t)n * DIM;
    for (int d = 0; d < DIM; ++d) nowE += fe[d] * wfs[d];
  }
  const float hisE = (hisIdx >= 0) ? ws[WS_EHIS + b * KH + hisIdx] : 0.f;
  const float val = (1.f - alpha) * db + alpha * (nowE + hisE) + ws[WS_CC + b];
  out[(size_t)b * NFIELDS + n] = val;
}

// ---------------------------------------------------------------------------
extern "C" void kernel_launch(void* const* d_in, const int* in_sizes, int n_in,
                              void* d_out, int out_size, void* d_ws,
                              size_t ws_size, hipStream_t stream) {
  (void)in_sizes; (void)n_in; (void)out_size; (void)ws_size;
  const float* company_emb     = (const float*)d_in[0];
  const float* field_emb       = (const float*)d_in[1];
  const float* raw_field_embed = (const float*)d_in[2];
  const int*   now_nodes       = (const int*)d_in[3];
  const int*   his_nodes       = (const int*)d_in[4];
  const int*   com_id          = (const int*)d_in[5];
  const float* company_table   = (const float*)d_in[6];
  const float* field_table     = (const float*)d_in[7];
  const float* Wp              = (const float*)d_in[8];
  const float* bp              = (const float*)d_in[9];
  const float* theta           = (const float*)d_in[10];
  const float* alpha_fields    = (const float*)d_in[11];
  const float* W1              = (const float*)d_in[12];
  const float* b1              = (const float*)d_in[13];
  const float* W2              = (const float*)d_in[14];
  const float* b2              = (const float*)d_in[15];
  const float* w_field         = (const float*)d_in[16];
  const float* b_field         = (const float*)d_in[17];
  const float* w_company       = (const float*)d_in[18];
  const float* b_company       = (const float*)d_in[19];
  float* out = (float*)d_out;
  float* ws  = (float*)d_ws;

  setup_kernel<<<1, 128, 0, stream>>>(company_emb, com_id, company_table, Wp,
                                      bp, theta, W2, b2, w_field, b_field,
                                      w_company, b_company, ws);

  mlp_wmma_kernel<<<MROWS / 16, 128, 0, stream>>>(raw_field_embed, his_nodes,
                                                  W1, b1, ws);

  const int wavesPerBlock = 256 / 32;
  const int ndBlocks = (NFIELDS + wavesPerBlock - 1) / wavesPerBlock;
  basedot_kernel<<<ndBlocks, 256, 0, stream>>>(field_table, ws);

  broadcast_kernel<<<(NFIELDS + 255) / 256, 256, 0, stream>>>(ws, out);

  correction_kernel<<<NB, 192, 0, stream>>>(field_emb, now_nodes, his_nodes,
                                            alpha_fields, w_field, ws, out);
}